// WeierstrassEllipticFunction_17635135717611
// MI455X (gfx1250) — compile-verified
//
#include <hip/hip_runtime.h>
#include <stdint.h>

// ---------------------------------------------------------------------------
// Weierstrass ℘ / ℘' on MI455X (gfx1250, wave32).
// Compute-bound FP32 VALU kernel; lattice constants staged to LDS via the
// CDNA5 Tensor Data Mover (tensor_load_to_lds + s_wait_tensorcnt).
// ---------------------------------------------------------------------------

#ifndef __has_builtin
#define __has_builtin(x) 0
#endif

#if __has_builtin(__builtin_amdgcn_tensor_load_to_lds) && __has_builtin(__builtin_amdgcn_s_wait_tensorcnt)
#define HAVE_TDM 1
#else
#define HAVE_TDM 0
#endif

#if defined(__has_include)
#if __has_include(<hip/amd_detail/amd_gfx1250_TDM.h>)
#define TDM_SIX_ARGS 1 // therock-10.0 headers present -> clang-23 6-arg builtin
#else
#define TDM_SIX_ARGS 0 // ROCm 7.2 / clang-22 5-arg builtin
#endif
#else
#define TDM_SIX_ARGS 0
#endif

typedef unsigned int u32x4 __attribute__((ext_vector_type(4)));
typedef int i32x8 __attribute__((ext_vector_type(8)));
typedef int i32x4 __attribute__((ext_vector_type(4)));

// ---------------------------------------------------------------------------
// Lattice table: (Wr, Wi, Ar, Ai) with A = 1/W^2, in the reference's
// stable-sorted order (sorted by |2mL + 2nLi|, ties in (m outer, n inner)
// iteration order, truncated to 64 points).
// ---------------------------------------------------------------------------
struct alignas(16) F4c {
  float wr, wi, ar, ai;
};
struct WTab {
  F4c v[64];
};

constexpr int kMN[64][2] = {
    // r^2 = 1
    {-1, 0}, {0, -1}, {0, 1}, {1, 0},
    // r^2 = 2
    {-1, -1}, {-1, 1}, {1, -1}, {1, 1},
    // r^2 = 4
    {-2, 0}, {0, -2}, {0, 2}, {2, 0},
    // r^2 = 5
    {-2, -1}, {-2, 1}, {-1, -2}, {-1, 2}, {1, -2}, {1, 2}, {2, -1}, {2, 1},
    // r^2 = 8
    {-2, -2}, {-2, 2}, {2, -2}, {2, 2},
    // r^2 = 9
    {-3, 0}, {0, -3}, {0, 3}, {3, 0},
    // r^2 = 10
    {-3, -1}, {-3, 1}, {-1, -3}, {-1, 3}, {1, -3}, {1, 3}, {3, -1}, {3, 1},
    // r^2 = 13
    {-3, -2}, {-3, 2}, {-2, -3}, {-2, 3}, {2, -3}, {2, 3}, {3, -2}, {3, 2},
    // r^2 = 16
    {-4, 0}, {0, -4}, {0, 4}, {4, 0},
    // r^2 = 17
    {-4, -1}, {-4, 1}, {-1, -4}, {-1, 4}, {1, -4}, {1, 4}, {4, -1}, {4, 1},
    // r^2 = 18
    {-3, -3}, {-3, 3}, {3, -3}, {3, 3},
    // r^2 = 20 (first 4 of 8 in stable order)
    {-4, -2}, {-4, 2}, {-2, -4}, {-2, 4}};

constexpr WTab makeTab() {
  WTab t{};
  const double L = 2.62205755429212;
  for (int i = 0; i < 64; ++i) {
    double wr = 2.0 * kMN[i][0] * L;
    double wi = 2.0 * kMN[i][1] * L;
    double r2 = wr * wr + wi * wi;
    double r4 = r2 * r2;
    t.v[i].wr = (float)wr;
    t.v[i].wi = (float)wi;
    t.v[i].ar = (float)((wr * wr - wi * wi) / r4); // Re 1/W^2
    t.v[i].ai = (float)((-2.0 * wr * wi) / r4);    // Im 1/W^2
  }
  return t;
}

__constant__ WTab gWTab = makeTab();

__device__ __forceinline__ float fast_rcp(float r) {
  // v_rcp_f32 (+/- ~1 ulp) + one Newton step -> ~full fp32 precision.
  float x = __builtin_amdgcn_rcpf(r);
  return x * (2.0f - r * x);
}

__device__ __forceinline__ float clampf(float v, float c) {
  return fminf(fmaxf(v, -c), c);
}

__global__ __launch_bounds__(256) void weierstrass_wp_kernel(
    const float* __restrict__ z_real, const float* __restrict__ z_imag,
    float* __restrict__ out, int N) {
  // Only __shared__ object in the kernel -> lives at LDS offset 0, which the
  // TDM descriptor's lds_addr field references directly.
  __shared__ F4c sW[64];

#if HAVE_TDM
  if ((threadIdx.x >> 5) == 0) { // wave 0 issues the DMA (EXEC ignored by TDM)
    uint64_t ga = (uint64_t)(uintptr_t)&gWTab.v[0];
    // D# group 0: count=1 (valid), lds_addr=0, global_addr, type=2 ("image")
    u32x4 g0;
    g0[0] = 1u;
    g0[1] = 0u;
    g0[2] = (unsigned)(ga & 0xFFFFFFFFu);
    g0[3] = (unsigned)((ga >> 32) & 0x01FFFFFFu) | 0x80000000u;
    // D# group 1: data_size=3 (8B units), tensor_dim0=128, tensor_dim1=1,
    // tile_dim0=128 (1-D tile: 128 x 8B = 1024B), dim0 stride=128.
    i32x8 g1;
    g1[0] = (int)(3u << 16);
    g1[1] = (int)(128u << 16);
    g1[2] = (int)(1u << 16);
    g1[3] = (int)(128u << 16);
    g1[4] = 0;
    g1[5] = 128;
    g1[6] = 0;
    g1[7] = 0;
    i32x4 gz = {0, 0, 0, 0}; // groups 2/3 unused (<=2D tensor)
#if TDM_SIX_ARGS
    i32x8 gz8 = {0, 0, 0, 0, 0, 0, 0, 0};
    __builtin_amdgcn_tensor_load_to_lds(g0, g1, gz, gz, gz8, 0);
#else
    __builtin_amdgcn_tensor_load_to_lds(g0, g1, gz, gz, 0);
#endif
    __builtin_amdgcn_s_wait_tensorcnt(0);
  }
#else
  if (threadIdx.x < 64) sW[threadIdx.x] = gWTab.v[threadIdx.x];
#endif
  __syncthreads();

  // The TDM engine wrote sW behind the compiler's back; without this, LLVM
  // sees no stores to sW, folds the ds loads to undef and deletes the series
  // loop entirely (observed in round 1: ds=0, valu=38). Escaping sW's address
  // into an asm with a "memory" clobber forces the loads to be performed,
  // while keeping them in addrspace(3) (ds_load_b128 broadcasts).
  asm volatile("" : : "v"(&sW[0]) : "memory");

  int i = blockIdx.x * blockDim.x + threadIdx.x;
  if (i >= N) return;

  const float zr = z_real[i];
  const float zi = z_imag[i];

  float swr = 0.0f, swi = 0.0f; // wp series accumulators
  float spr = 0.0f, spi = 0.0f; // wp' series accumulators

#pragma unroll 4
  for (int k = 0; k < 64; ++k) {
    const F4c wv = sW[k]; // broadcast ds_load_b128, conflict-free

    const float dr = zr - wv.wr;
    const float di = zi - wv.wi;
    const float dr2 = dr * dr;
    const float di2 = di * di;
    const float r2 = dr2 + di2;

    // mask: 2e-7 < |d| < 1000  <=>  4e-14 < |d|^2 < 1e6
    const bool m = (r2 > 4.0e-14f) && (r2 < 1.0e6f);

    const float inv = fast_rcp(r2); // 1/|d|^2
    const float ir4 = inv * inv;    // 1/|d|^4

    // 1/d^2 = conj(d)^2 / |d|^4
    const float c2r = dr2 - di2;
    const float c2i = -2.0f * dr * di;
    const float i2r = c2r * ir4;
    const float i2i = c2i * ir4;

    const float wr_t = clampf(i2r - wv.ar, 1000.0f);
    const float wi_t = clampf(i2i - wv.ai, 1000.0f);

    // -2/d^3 = (1/d^2) * (-2 conj(d) / |d|^2); bounded, no Inf where ref finite
    const float m2x = -2.0f * inv;
    const float qr = dr * m2x;
    const float qi = -di * m2x;
    const float pr_t = clampf(i2r * qr - i2i * qi, 1000.0f);
    const float pi_t = clampf(i2r * qi + i2i * qr, 1000.0f);

    swr += m ? wr_t : 0.0f;
    swi += m ? wi_t : 0.0f;
    spr += m ? pr_t : 0.0f;
    spi += m ? pi_t : 0.0f;
  }

  // ---- main terms 1/z^2 and -2/z^3 with stability masking ----
  const float r2z = zr * zr + zi * zi;
  const bool stable = (r2z > 4.0e-14f) && (r2z < 1.0e4f);
  const float zrs = stable ? zr : 1.0f;
  const float zis = stable ? zi : 0.0f;
  const float r2s = stable ? r2z : 1.0f;

  const float inv = fast_rcp(r2s);
  const float ir4 = inv * inv;
  const float c2r = zrs * zrs - zis * zis;
  const float c2i = -2.0f * zrs * zis;
  const float wmr = c2r * ir4;
  const float wmi = c2i * ir4;
  const float m2x = -2.0f * inv;
  const float qr = zrs * m2x;
  const float qi = -zis * m2x;
  const float pmr = wmr * qr - wmi * qi;
  const float pmi = wmr * qi + wmi * qr;

  float wpr = stable ? (wmr + swr) : 0.0f;
  float wpi = stable ? (wmi + swi) : 0.0f;
  float ppr = stable ? (pmr + spr) : 0.0f;
  float ppi = stable ? (pmi + spi) : 0.0f;

  // invalid = near_origin | !isfinite; all computed paths above are bounded
  // (series clamped, main term <= ~2.5e20), so only near_origin can fire.
  const bool invalid = r2z < 4.0e-14f;
  wpr = invalid ? 100.0f : wpr;
  wpi = invalid ? 0.0f : wpi;
  ppr = invalid ? 100.0f : ppr;
  ppi = invalid ? 0.0f : ppi;

  float2 wp, wpp;
  wp.x = clampf(wpr, 5000.0f);
  wp.y = clampf(wpi, 5000.0f);
  wpp.x = clampf(ppr, 5000.0f);
  wpp.y = clampf(ppi, 5000.0f);

  // Output layout: wp[N] complex, then wpp[N] complex (interleaved re,im).
  float2* o = (float2*)out;
  o[i] = wp;
  o[N + i] = wpp;
}

extern "C" void kernel_launch(void* const* d_in, const int* in_sizes, int n_in,
                              void* d_out, int out_size, void* d_ws,
                              size_t ws_size, hipStream_t stream) {
  (void)n_in;
  (void)out_size;
  (void)d_ws;
  (void)ws_size;
  const float* z_real = (const float*)d_in[0];
  const float* z_imag = (const float*)d_in[1];
  float* out = (float*)d_out;
  const int N = in_sizes[0]; // 4*2048*64 = 524288

  const int block = 256; // 8 wave32 waves per block
  const int grid = (N + block - 1) / block;
  weierstrass_wp_kernel<<<grid, block, 0, stream>>>(z_real, z_imag, out, N);
}